// Attention_81140522156924
// MI455X (gfx1250) — compile-verified
//
#include <hip/hip_runtime.h>
#include <hip/hip_bf16.h>

typedef float v2f __attribute__((ext_vector_type(2)));
typedef float v8f __attribute__((ext_vector_type(8)));
typedef unsigned int u32x4 __attribute__((ext_vector_type(4)));
typedef int i32x8 __attribute__((ext_vector_type(8)));
typedef int i32x4 __attribute__((ext_vector_type(4)));

#define BATCH 4
#define SEQ 2048
#define NH 8
#define DH 48
#define DMODEL 384
#define QKVW (3 * DMODEL)                 // 1152
#define SCALE_ATTN 0.14433756729740643f   // 48^-0.5

#define GKC 32                            // GEMM K chunk
#define BS_PAD 17                         // TDM pad dwords per row
#define BS_STRIDE (64 + BS_PAD)           // 81: conflict-free frag reads

#define KV_TILE 64
#define LDS_KSTRIDE 52
#define LDS_PSTRIDE 66

__device__ __forceinline__ v8f wmma4(v2f a, v2f b, v8f c) {
  // D = A(16x4 f32) * B(4x16 f32) + C(16x16 f32)
  return __builtin_amdgcn_wmma_f32_16x16x4_f32(false, a, false, b, (short)0, c,
                                               false, false);
}

// ---------------------------------------------------------------------------
// TDM: async-copy a 32-row x 64-col f32 tile (row stride = nstride elements)
// from global memory into LDS at byte offset lds_addr, inserting 17 pad
// dwords after every 64-dword row (ISA D#: pad_interval=5, pad_amount=16).
// D# per CDNA5 ISA 8.3-8.6: count=1, type=2, data_size=4B, 2-D tile.
// 6-arg builtin form (clang-23 / therock-10.0 headers).
// ---------------------------------------------------------------------------
__device__ __forceinline__ void tdm_load_tile(const float* gsrc,
                                              unsigned lds_addr, int nstride) {
  const unsigned long long ga = (unsigned long long)(uintptr_t)gsrc;
  u32x4 g0 = {1u,                                   // count = 1 valid D#
              lds_addr,                             // lds_addr [63:32]
              (unsigned)ga,                         // global_addr lo
              (unsigned)((ga >> 32) & 0x01FFFFFFull) | 0x80000000u}; // +type=2
  i32x8 g1 = {(int)((2u << 16) |                    // data_size = 4B
                    (1u << 20) |                    // pad_enable
                    (5u << 22) |                    // pad_interval: 64 dwords
                    (16u << 25)),                   // pad_amount: 17 dwords
              (int)(64u << 16),                     // tensor_dim0 = 64 (lo16)
              (int)(32u << 16),                     // tensor_dim1 = 32 (lo16)
              (int)(64u << 16),                     // tile_dim0 = 64
              32,                                   // tile_dim1 = 32
              nstride,                              // tensor_dim0_stride lo32
              0, 0};
  i32x4 z4 = {0, 0, 0, 0};
  i32x8 z8 = {0, 0, 0, 0, 0, 0, 0, 0};
  __builtin_amdgcn_tensor_load_to_lds(g0, g1, z4, z4, z8, 0);
}

// ---------------------------------------------------------------------------
// fp32 WMMA GEMM: C[M,N] = A[M,K] @ B[K,N] (+ bias[N]).
// Block (8 waves) -> 128(M) x 64(N); B panel staged by TDM in K-chunks of 32,
// double-buffered in LDS; each wave computes 16x64 (4 accumulators).
// ---------------------------------------------------------------------------
__global__ __launch_bounds__(256) void gemm_wmma(const float* __restrict__ A,
                                                 const float* __restrict__ B,
                                                 const float* __restrict__ bias,
                                                 float* __restrict__ C,
                                                 int M, int N, int K) {
  __shared__ float Bs[2][GKC * BS_STRIDE];   // 2 x 10368 B

  const int lane = threadIdx.x & 31;
  const int wave = threadIdx.x >> 5;
  const int half = lane >> 4;
  const int l16  = lane & 15;

  const int ncols = N >> 6;
  const int mt = blockIdx.x / ncols;
  const int nc = blockIdx.x % ncols;
  const int row0 = mt * 128 + wave * 16;
  const int col0 = nc * 64;

  v8f acc[4] = {};

  // A frag: lane holds A[row0+l16][k + 2*half + j], j=0,1 (b64 loads)
  const float* arow = A + (size_t)(row0 + l16) * K + 2 * half;

  const int nk = K / GKC;
  if (wave == 0)
    tdm_load_tile(B + (size_t)0 * N + col0,
                  (unsigned)(uintptr_t)&Bs[0][0], N);

  for (int kc = 0; kc < nk; ++kc) {
    if (wave == 0) __builtin_amdgcn_s_wait_tensorcnt(0);  // chunk kc arrived
    __syncthreads();                                      // publish to block
    if (wave == 0 && kc + 1 < nk)
      tdm_load_tile(B + (size_t)((kc + 1) * GKC) * N + col0,
                    (unsigned)(uintptr_t)&Bs[(kc + 1) & 1][0], N);

    // Hoist this chunk's 8 A-fragments (one loadcnt wait total).
    v2f af[8];
#pragma unroll
    for (int i = 0; i < 8; ++i) {
      af[i].x = arow[kc * GKC + 4 * i + 0];
      af[i].y = arow[kc * GKC + 4 * i + 1];
    }

    const float* bs = &Bs[kc & 1][0];
#pragma unroll
    for (int i = 0; i < 8; ++i) {
#pragma unroll
      for (int t = 0; t < 4; ++t) {
        // B frag: B[kk][col], kk = 4i + 2*half + j  -> conflict-free banks
        const float* bp = bs + (4 * i + 2 * half) * BS_STRIDE + t * 16 + l16;
        v2f bf;
        bf.x = bp[0];
        bf.y = bp[BS_STRIDE];
        acc[t] = wmma4(af[i], bf, acc[t]);
      }
    }
    __syncthreads();   // all waves done with buf (overwritten at kc+2)
  }

  // C/D layout: lane l, VGPR r -> row r + 8*half, col l16 (+16t)
#pragma unroll
  for (int t = 0; t < 4; ++t) {
    const int col = col0 + t * 16 + l16;
    const float bv = bias ? bias[col] : 0.0f;
#pragma unroll
    for (int r = 0; r < 8; ++r) {
      const int row = row0 + r + 8 * half;
      C[(size_t)row * N + col] = acc[t][r] + bv;
    }
  }
}

// ---------------------------------------------------------------------------
// Flash attention over qkv[B,N,3,h,dh] (row-major [8192 x 1152]).
// Block = 4 waves; each wave owns 16 q-rows; KV tiled 64 rows at a time.
// ---------------------------------------------------------------------------
__global__ __launch_bounds__(128) void flash_attn(const float* __restrict__ qkv,
                                                  float* __restrict__ o_out) {
  __shared__ float Ks[KV_TILE * LDS_KSTRIDE];
  __shared__ float Vs[KV_TILE * LDS_KSTRIDE];
  __shared__ float Ps[4 * 16 * LDS_PSTRIDE];

  const int tid  = threadIdx.x;
  const int lane = tid & 31;
  const int wave = tid >> 5;
  const int half = lane >> 4;
  const int l16  = lane & 15;

  const int qb   = blockIdx.x & 31;
  const int bh   = blockIdx.x >> 5;
  const int head = bh & 7;
  const int b    = bh >> 3;
  const int q0   = qb * 64 + wave * 16;

  // Preload Q tile (16 x 48) as 12 A-fragments.
  const float* qrow =
      qkv + (size_t)(b * SEQ + q0 + l16) * QKVW + head * DH + 2 * half;
  v2f qf[12];
#pragma unroll
  for (int kb = 0; kb < 12; ++kb) {
    qf[kb].x = qrow[4 * kb + 0];
    qf[kb].y = qrow[4 * kb + 1];
  }

  v8f o0 = {}, o1 = {}, o2 = {};
  float mrow[8], lrow[8];
#pragma unroll
  for (int r = 0; r < 8; ++r) { mrow[r] = -1e30f; lrow[r] = 0.0f; }

  float* Pw = Ps + wave * 16 * LDS_PSTRIDE;
  const float* kbase = qkv + (size_t)(b * SEQ) * QKVW + DMODEL + head * DH;
  const float* vbase = kbase + DMODEL;

  for (int kv0 = 0; kv0 < SEQ; kv0 += KV_TILE) {
    // Cooperative K/V tile load: 64x48 each = 768 float4; 6 per thread.
#pragma unroll
    for (int i = 0; i < 6; ++i) {
      const int idx = i * 128 + tid;
      const int row = idx / 12;
      const int c4  = idx % 12;
      const size_t goff = (size_t)(kv0 + row) * QKVW + c4 * 4;
      *(float4*)(Ks + row * LDS_KSTRIDE + c4 * 4) = *(const float4*)(kbase + goff);
      *(float4*)(Vs + row * LDS_KSTRIDE + c4 * 4) = *(const float4*)(vbase + goff);
    }
    __syncthreads();

    // S = Q @ K^T (16 x 64): K rows read in A-layout give the K^T B-frag.
    v8f s[4] = {};
#pragma unroll
    for (int kb = 0; kb < 12; ++kb) {
#pragma unroll
      for (int t = 0; t < 4; ++t) {
        const float* kp = Ks + (t * 16 + l16) * LDS_KSTRIDE + 4 * kb + 2 * half;
        v2f bf;
        bf.x = kp[0];
        bf.y = kp[1];
        s[t] = wmma4(qf[kb], bf, s[t]);
      }
    }
#pragma unroll
    for (int t = 0; t < 4; ++t) s[t] *= SCALE_ATTN;

    // Online softmax. Lane holds rows r+8*half, cols l16+16t.
#pragma unroll
    for (int r = 0; r < 8; ++r) {
      float mx = fmaxf(fmaxf(s[0][r], s[1][r]), fmaxf(s[2][r], s[3][r]));
#pragma unroll
      for (int off = 1; off < 16; off <<= 1)
        mx = fmaxf(mx, __shfl_xor(mx, off, 32));
      const float mnew = fmaxf(mrow[r], mx);
      const float corr = __expf(mrow[r] - mnew);
      mrow[r] = mnew;
      float rs = 0.0f;
#pragma unroll
      for (int t = 0; t < 4; ++t) {
        const float p = __expf(s[t][r] - mnew);
        s[t][r] = p;
        rs += p;
      }
#pragma unroll
      for (int off = 1; off < 16; off <<= 1)
        rs += __shfl_xor(rs, off, 32);
      lrow[r] = lrow[r] * corr + rs;
      o0[r] *= corr;
      o1[r] *= corr;
      o2[r] *= corr;
    }

    // Stage P (C-layout) into per-wave LDS so it can be re-read in A-layout.
#pragma unroll
    for (int t = 0; t < 4; ++t)
#pragma unroll
      for (int r = 0; r < 8; ++r)
        Pw[(r + 8 * half) * LDS_PSTRIDE + t * 16 + l16] = s[t][r];

    // Wave-local LDS RAW: DS ops are in-order per wave; drain DScnt only.
    asm volatile("s_wait_dscnt 0" ::: "memory");

    // O += P(16x64) @ V(64x48): 16 K-steps, 3 output subtiles.
#pragma unroll
    for (int kb = 0; kb < 16; ++kb) {
      const float* pp = Pw + l16 * LDS_PSTRIDE + 4 * kb + 2 * half;
      v2f af;
      af.x = pp[0];
      af.y = pp[1];
      const float* vp = Vs + (4 * kb + 2 * half) * LDS_KSTRIDE + l16;
      v2f b0, b1, b2;
      b0.x = vp[0];  b0.y = vp[LDS_KSTRIDE];
      b1.x = vp[16]; b1.y = vp[LDS_KSTRIDE + 16];
      b2.x = vp[32]; b2.y = vp[LDS_KSTRIDE + 32];
      o0 = wmma4(af, b0, o0);
      o1 = wmma4(af, b1, o1);
      o2 = wmma4(af, b2, o2);
    }
    __syncthreads();  // everyone done with Ks/Vs before next tile overwrites
  }

  // Normalize and write [B, N, h*dh] slice.
#pragma unroll
  for (int r = 0; r < 8; ++r) {
    const int row = r + 8 * half;
    const float inv = 1.0f / lrow[r];
    float* op =
        o_out + (size_t)(b * SEQ + q0 + row) * DMODEL + head * DH + l16;
    op[0]  = o0[r] * inv;
    op[16] = o1[r] * inv;
    op[32] = o2[r] * inv;
  }
}

// ---------------------------------------------------------------------------
extern "C" void kernel_launch(void* const* d_in, const int* in_sizes, int n_in,
                              void* d_out, int out_size, void* d_ws,
                              size_t ws_size, hipStream_t stream) {
  const float* x     = (const float*)d_in[0];  // [4,2048,384]
  const float* Wqkv  = (const float*)d_in[1];  // [384,1152]
  const float* Wproj = (const float*)d_in[2];  // [384,384]
  const float* bproj = (const float*)d_in[3];  // [384]
  float* out = (float*)d_out;                  // [4,2048,384]

  const int M = BATCH * SEQ;                   // 8192
  float* qkv  = (float*)d_ws;                  // [8192 x 1152]
  float* attn = qkv + (size_t)M * QKVW;        // [8192 x 384]

  // 1) QKV projection: [8192,384] @ [384,1152]
  {
    const int blocks = (M / 128) * (QKVW / 64);  // 64 * 18 = 1152
    gemm_wmma<<<blocks, 256, 0, stream>>>(x, Wqkv, nullptr, qkv, M, QKVW,
                                          DMODEL);
  }

  // 2) Flash attention: grid = B * h * (N/64) = 1024 blocks of 128 threads
  flash_attn<<<BATCH * NH * (SEQ / 64), 128, 0, stream>>>(qkv, attn);

  // 3) Output projection + bias: [8192,384] @ [384,384]
  {
    const int blocks = (M / 128) * (DMODEL / 64);  // 64 * 6 = 384
    gemm_wmma<<<blocks, 256, 0, stream>>>(attn, Wproj, bproj, out, M, DMODEL,
                                          DMODEL);
  }
}